// MyOURS_CRAZY_7713761263762
// MI455X (gfx1250) — compile-verified
//
#include <hip/hip_runtime.h>
#include <math.h>

// Problem constants (match reference)
#define Bsz 16
#define Pn  12
#define Qn  12
#define Nn  1024
#define Dn  64
#define Hn  32
#define MT  (Bsz * Nn)          // 16384  (rows of the recurrent state)
#define BPN (Bsz * Pn * Nn)     // 196608 (rows of the embedded input)

typedef __attribute__((ext_vector_type(2))) float v2f;
typedef __attribute__((ext_vector_type(8))) float v8f;

// CDNA5 full-precision f32 WMMA: D(16x16,f32) = A(16x4,f32) * B(4x16,f32) + C
__device__ __forceinline__ v8f wmma4(v2f a, v2f b, v8f c) {
  return __builtin_amdgcn_wmma_f32_16x16x4_f32(
      /*neg_a=*/false, a, /*neg_b=*/false, b,
      /*c_mod=*/(short)0, c, /*reuse_a=*/false, /*reuse_b=*/false);
}

__device__ __forceinline__ float actf(float x, int act) {
  if (act == 1) return x > 0.f ? x : 0.f;          // relu
  if (act == 2) return 1.f / (1.f + expf(-x));     // sigmoid
  if (act == 3) return tanhf(x);                   // tanh
  return x;
}

// ---------------------------------------------------------------------------
// Chebyshev diffusion, channel-major activations.
// XDT layout: channel-major (512 channels, MT) where channel = slot*128 + c.
//   acc[m,c] = sum_n L[m,n] * XDT[src*128+c][b*1024+n]
//   XDT[dst*128+c][b*1024+m] = prev<0 ? acc : 2*acc - XDT[prev*128+c][...]
// One wave -> 16x64 tile. B-fragment K-pair is contiguous -> single b64 load.
// ---------------------------------------------------------------------------
__global__ void diffuse_t(const float* __restrict__ L,
                          float* __restrict__ XDT,
                          int src, int dst, int prev) {
  const int lane = threadIdx.x;
  const int lr = lane & 15;          // A row-in-tile / B column
  const int hi = lane >> 4;          // K sub-pair select (ISA f32 16x16x4 layout)
  const int m0 = blockIdx.x << 4;    // 0..1008
  const int c0 = blockIdx.y << 6;    // 0 or 64
  const int b  = blockIdx.z;

  const float* __restrict__ Lr = L + (size_t)(m0 + lr) * Nn + 2 * hi;
  const float* __restrict__ Xs =
      XDT + (size_t)(src * 128 + c0 + lr) * MT + (size_t)b * Nn + 2 * hi;

  v8f a0 = {}, a1 = {}, a2 = {}, a3 = {};
#pragma unroll 4
  for (int k = 0; k < Nn; k += 4) {
    v2f av = { Lr[k], Lr[k + 1] };
    v2f b0 = { Xs[k],                       Xs[k + 1] };
    v2f b1 = { Xs[k + (size_t)16 * MT],     Xs[k + 1 + (size_t)16 * MT] };
    v2f b2 = { Xs[k + (size_t)32 * MT],     Xs[k + 1 + (size_t)32 * MT] };
    v2f b3 = { Xs[k + (size_t)48 * MT],     Xs[k + 1 + (size_t)48 * MT] };
    a0 = wmma4(av, b0, a0);
    a1 = wmma4(av, b1, a1);
    a2 = wmma4(av, b2, a2);
    a3 = wmma4(av, b3, a3);
  }

  // epilogue: lane writes 8 consecutive floats per accumulator (coalesced)
  float* Xo = XDT + (size_t)(dst * 128 + c0 + lr) * MT + (size_t)b * Nn + m0 + 8 * hi;
  if (prev >= 0) {
    const float* Xp =
        XDT + (size_t)(prev * 128 + c0 + lr) * MT + (size_t)b * Nn + m0 + 8 * hi;
#pragma unroll
    for (int i = 0; i < 8; ++i) {
      Xo[i]                      = 2.f * a0[i] - Xp[i];
      Xo[i + (size_t)16 * MT]    = 2.f * a1[i] - Xp[i + (size_t)16 * MT];
      Xo[i + (size_t)32 * MT]    = 2.f * a2[i] - Xp[i + (size_t)32 * MT];
      Xo[i + (size_t)48 * MT]    = 2.f * a3[i] - Xp[i + (size_t)48 * MT];
    }
  } else {
#pragma unroll
    for (int i = 0; i < 8; ++i) {
      Xo[i]                   = a0[i];
      Xo[i + (size_t)16 * MT] = a1[i];
      Xo[i + (size_t)32 * MT] = a2[i];
      Xo[i + (size_t)48 * MT] = a3[i];
    }
  }
}

// ---------------------------------------------------------------------------
// GEMM, transposed (channel-major) A and output:
//   OutT[c][m] = act( sum_k AT[k][m] * Bw[k*ldb + c] + bias[c] )
// AT is (K, Mtot) with leading dim ldat. One wave -> 16x32 tile.
// K must be a multiple of 4. grid.x = M/16, grid.y = Nc/32.
// ---------------------------------------------------------------------------
__global__ void gemm_t(const float* __restrict__ AT, int ldat,
                       const float* __restrict__ Bw, int ldb,
                       const float* __restrict__ bias,
                       float* __restrict__ OutT, int ldot,
                       int K, int act) {
  const int lane = threadIdx.x;
  const int lr = lane & 15;
  const int hi = lane >> 4;
  const int m0 = blockIdx.x << 4;
  const int c0 = blockIdx.y << 5;

  const float* Ac = AT + (size_t)(2 * hi) * ldat + m0 + lr;   // element (m0+lr, 2hi)
  v8f a0 = {}, a1 = {};
#pragma unroll 4
  for (int k = 0; k < K; k += 4) {
    v2f av = { Ac[(size_t)k * ldat], Ac[(size_t)(k + 1) * ldat] };
    const float* B0 = Bw + (size_t)(k + 2 * hi) * ldb + c0 + lr;
    v2f b0 = { B0[0],  B0[ldb] };
    v2f b1 = { B0[16], B0[16 + ldb] };
    a0 = wmma4(av, b0, a0);
    a1 = wmma4(av, b1, a1);
  }
  const float bi0 = bias[c0 + lr];
  const float bi1 = bias[c0 + 16 + lr];
  float* O0 = OutT + (size_t)(c0 + lr) * ldot + m0 + 8 * hi;
  float* O1 = OutT + (size_t)(c0 + 16 + lr) * ldot + m0 + 8 * hi;
#pragma unroll
  for (int i = 0; i < 8; ++i) {
    O0[i] = actf(a0[i] + bi0, act);
    O1[i] = actf(a1[i] + bi1, act);
  }
}

// ---------------------------------------------------------------------------
// GEMM, row-major A (contiguous K per row), transposed output. Guarded K tail
// (K=98 for the ZC patch-contraction). One wave -> 16x32 tile.
// ---------------------------------------------------------------------------
__global__ void gemm_rm(const float* __restrict__ A, int lda,
                        const float* __restrict__ Bw, int ldb,
                        const float* __restrict__ bias,
                        float* __restrict__ OutT, int ldot,
                        int K, int act) {
  const int lane = threadIdx.x;
  const int lr = lane & 15;
  const int hi = lane >> 4;
  const int m0 = blockIdx.x << 4;
  const int c0 = blockIdx.y << 5;

  const float* Ar = A + (size_t)(m0 + lr) * lda + 2 * hi;
  v8f a0 = {}, a1 = {};
  const int Kf = K & ~3;
#pragma unroll 4
  for (int k = 0; k < Kf; k += 4) {
    v2f av = { Ar[k], Ar[k + 1] };
    const float* B0 = Bw + (size_t)(k + 2 * hi) * ldb + c0 + lr;
    v2f b0 = { B0[0],  B0[ldb] };
    v2f b1 = { B0[16], B0[16 + ldb] };
    a0 = wmma4(av, b0, a0);
    a1 = wmma4(av, b1, a1);
  }
  if (Kf < K) {
    const int k0 = Kf + 2 * hi, k1 = k0 + 1;
    v2f av = { (k0 < K) ? Ar[Kf] : 0.f, (k1 < K) ? Ar[Kf + 1] : 0.f };
    v2f b0 = { (k0 < K) ? Bw[(size_t)k0 * ldb + c0 + lr] : 0.f,
               (k1 < K) ? Bw[(size_t)k1 * ldb + c0 + lr] : 0.f };
    v2f b1 = { (k0 < K) ? Bw[(size_t)k0 * ldb + c0 + 16 + lr] : 0.f,
               (k1 < K) ? Bw[(size_t)k1 * ldb + c0 + 16 + lr] : 0.f };
    a0 = wmma4(av, b0, a0);
    a1 = wmma4(av, b1, a1);
  }
  const float bi0 = bias[c0 + lr];
  const float bi1 = bias[c0 + 16 + lr];
  float* O0 = OutT + (size_t)(c0 + lr) * ldot + m0 + 8 * hi;
  float* O1 = OutT + (size_t)(c0 + 16 + lr) * ldot + m0 + 8 * hi;
#pragma unroll
  for (int i = 0; i < 8; ++i) {
    O0[i] = actf(a0[i] + bi0, act);
    O1[i] = actf(a1[i] + bi1, act);
  }
}

// ------------------------- small VALU kernels ------------------------------

// rows x 32 @ 32 x 32 (+bias, act), row-major small buffers
__global__ void mm32_kernel(const float* __restrict__ in, const float* __restrict__ W,
                            const float* __restrict__ bias, float* __restrict__ out,
                            int rows, int act) {
  int id = blockIdx.x * blockDim.x + threadIdx.x;
  if (id >= rows * 32) return;
  int n = id >> 5, h = id & 31;
  float acc = bias[h];
#pragma unroll
  for (int k = 0; k < 32; ++k) acc += in[n * 32 + k] * W[k * 32 + h];
  out[id] = actf(acc, act);
}

// one-hot(day,7)||one-hot(tod,288) @ W_te1 == W_te1[day] + W_te1[7+tod]; +bias, relu
__global__ void te_gather(const int* __restrict__ TE, const float* __restrict__ W_te1,
                          const float* __restrict__ b_te1, float* __restrict__ out) {
  int id = blockIdx.x * blockDim.x + threadIdx.x;
  if (id >= Bsz * (Pn + Qn) * Hn) return;
  int h = id & 31, r = id >> 5;
  int day = TE[r * 2 + 0];
  int tod = TE[r * 2 + 1];
  float v = W_te1[day * 32 + h] + W_te1[(7 + tod) * 32 + h] + b_te1[h];
  out[id] = v > 0.f ? v : 0.f;
}

// XEMBT[h][m] = relu(X[m]*W_in1+b_in1) @ W_in2[:,h] + b_in2[h] + se[n,h] + te[b,p,h]
// channel-major output, channel = blockIdx.y (0..31); m over B*P*N.
__global__ void input_embed_t(const float* __restrict__ X,
                              const float* __restrict__ W_in1, const float* __restrict__ b_in1,
                              const float* __restrict__ W_in2, const float* __restrict__ b_in2,
                              const float* __restrict__ seO, const float* __restrict__ teO,
                              float* __restrict__ XEMBT) {
  int m = blockIdx.x * blockDim.x + threadIdx.x;
  int h = blockIdx.y;
  if (m >= BPN) return;
  int n = m % Nn;
  int bp = m / Nn;
  int p = bp % Pn;
  int b = bp / Pn;
  float xv = X[m];
  float acc = b_in2[h];
#pragma unroll
  for (int d = 0; d < 64; ++d) {
    float t = xv * W_in1[d] + b_in1[d];
    t = t > 0.f ? t : 0.f;
    acc += t * W_in2[d * 32 + h];
  }
  acc += seO[n * 32 + h] + teO[(b * (Pn + Qn) + p) * 32 + h];
  XEMBT[(size_t)h * BPN + m] = acc;
}

// XDT slot0 = [x_t | h]  (channel-major); channel c = blockIdx.y (0..127)
__global__ void build_cat0_t(float* __restrict__ XDT, const float* __restrict__ XEMBT,
                             const float* __restrict__ HT, int t) {
  int m = blockIdx.x * blockDim.x + threadIdx.x;
  int c = blockIdx.y;
  if (m >= MT) return;
  int b = m >> 10, n = m & (Nn - 1);
  float v;
  if (c < 64) v = XEMBT[(size_t)c * BPN + (size_t)(b * Pn + t) * Nn + n];
  else        v = HT[(size_t)(c - 64) * MT + m];
  XDT[(size_t)c * MT + m] = v;
}

// XDT slot0 channels 64..127 = r * h   (r = RUT channels 0..63)
__global__ void build_cat1_t(float* __restrict__ XDT, const float* __restrict__ RUT,
                             const float* __restrict__ HT) {
  int m = blockIdx.x * blockDim.x + threadIdx.x;
  int j = blockIdx.y;   // 0..63
  if (m >= MT) return;
  XDT[(size_t)(64 + j) * MT + m] = RUT[(size_t)j * MT + m] * HT[(size_t)j * MT + m];
}

// h = u*h + (1-u)*c   (u = RUT channels 64..127)
__global__ void update_h_t(float* __restrict__ HT, const float* __restrict__ RUT,
                           const float* __restrict__ CCT) {
  int m = blockIdx.x * blockDim.x + threadIdx.x;
  int j = blockIdx.y;   // 0..63
  if (m >= MT) return;
  float u = RUT[(size_t)(64 + j) * MT + m];
  HT[(size_t)j * MT + m] = u * HT[(size_t)j * MT + m] + (1.f - u) * CCT[(size_t)j * MT + m];
}

__global__ void zero_kernel(float* __restrict__ p, int n) {
  int id = blockIdx.x * blockDim.x + threadIdx.x;
  if (id < n) p[id] = 0.f;
}

// out[b,q,n] = Y1T[:, m] @ W_o2[:,q] + b_o2[q];  id = q*MT + m
__global__ void out_proj_t(const float* __restrict__ Y1T, const float* __restrict__ W_o2,
                           const float* __restrict__ b_o2, float* __restrict__ out) {
  int id = blockIdx.x * blockDim.x + threadIdx.x;
  if (id >= Qn * MT) return;
  int m = id & (MT - 1);
  int q = id >> 14;
  int b = m >> 10, n = m & (Nn - 1);
  float acc = b_o2[q];
#pragma unroll
  for (int d = 0; d < 64; ++d)
    acc += Y1T[(size_t)d * MT + m] * W_o2[d * Qn + q];
  out[(size_t)b * (Qn * Nn) + q * Nn + n] = acc;
}

// ---------------------------------------------------------------------------

extern "C" void kernel_launch(void* const* d_in, const int* in_sizes, int n_in,
                              void* d_out, int out_size, void* d_ws, size_t ws_size,
                              hipStream_t stream) {
  (void)in_sizes; (void)n_in; (void)out_size; (void)ws_size;

  const float* X      = (const float*)d_in[0];
  const float* ZC     = (const float*)d_in[1];
  const int*   TE     = (const int*)  d_in[2];
  const float* Ls     = (const float*)d_in[3];
  const float* SE     = (const float*)d_in[4];
  const float* W_se1  = (const float*)d_in[5];
  const float* b_se1  = (const float*)d_in[6];
  const float* W_se2  = (const float*)d_in[7];
  const float* b_se2  = (const float*)d_in[8];
  const float* W_te1  = (const float*)d_in[9];
  const float* b_te1  = (const float*)d_in[10];
  const float* W_te2  = (const float*)d_in[11];
  const float* b_te2  = (const float*)d_in[12];
  const float* W_in1  = (const float*)d_in[13];
  const float* b_in1  = (const float*)d_in[14];
  const float* W_in2  = (const float*)d_in[15];
  const float* b_in2  = (const float*)d_in[16];
  const float* W_conv = (const float*)d_in[17];
  const float* b_conv = (const float*)d_in[18];
  const float* W_g    = (const float*)d_in[19];
  const float* b_g    = (const float*)d_in[20];
  const float* W_c    = (const float*)d_in[21];
  const float* b_c    = (const float*)d_in[22];
  const float* W_o1   = (const float*)d_in[23];
  const float* b_o1   = (const float*)d_in[24];
  const float* W_o2   = (const float*)d_in[25];
  const float* b_o2   = (const float*)d_in[26];
  float* out = (float*)d_out;

  // workspace carve-up (floats); total ~26.3M floats (~105 MB)
  float* ws    = (float*)d_ws;
  float* SE_T  = ws;                   // 32768
  float* SE_O  = SE_T + 32768;         // 32768
  float* TE_T  = SE_O + 32768;         // 12288
  float* TE_O  = TE_T + 12288;         // 12288
  float* XEMBT = TE_O + 12288;         // 64 * BPN = 12582912 (channel-major)
  float* XDT   = XEMBT + 12582912;     // 512 * MT = 8388608  (channel-major)
  float* HT    = XDT + 8388608;        // 64 * MT  = 1048576
  float* RUT   = HT + 1048576;         // 128 * MT = 2097152
  float* CCT   = RUT + 2097152;        // 64 * MT  = 1048576
  float* Y1T   = CCT + 1048576;        // 64 * MT  = 1048576

  const int TB = 256;

  // --- spatial embedding: se = relu(SE@W_se1+b) @ W_se2 + b ---
  mm32_kernel<<<(Nn * 32 + TB - 1) / TB, TB, 0, stream>>>(SE, W_se1, b_se1, SE_T, Nn, 1);
  mm32_kernel<<<(Nn * 32 + TB - 1) / TB, TB, 0, stream>>>(SE_T, W_se2, b_se2, SE_O, Nn, 0);

  // --- temporal embedding (one-hot collapses to row gather) ---
  te_gather<<<(Bsz * 24 * 32 + TB - 1) / TB, TB, 0, stream>>>(TE, W_te1, b_te1, TE_T);
  mm32_kernel<<<(Bsz * 24 * 32 + TB - 1) / TB, TB, 0, stream>>>(TE_T, W_te2, b_te2, TE_O,
                                                                Bsz * 24, 0);

  // --- ZC conv == GEMM (196608 x 98) @ (98 x 32), relu -> XEMBT channels 32..63 ---
  gemm_rm<<<dim3(BPN / 16, 1), 32, 0, stream>>>(
      ZC, 98, W_conv, 32, b_conv, XEMBT + (size_t)32 * BPN, BPN, 98, 1);

  // --- input embedding + STE add -> XEMBT channels 0..31 ---
  input_embed_t<<<dim3((BPN + TB - 1) / TB, 32), TB, 0, stream>>>(
      X, W_in1, b_in1, W_in2, b_in2, SE_O, TE_O, XEMBT);

  // --- h0 = 0 ---
  zero_kernel<<<(64 * MT + TB - 1) / TB, TB, 0, stream>>>(HT, 64 * MT);

  const dim3 dgrid(Nn / 16, 2, Bsz);   // 16x64 tiles, 128 channels, 16 batches
  const dim3 egrid0(MT / TB, 128), egrid1(MT / TB, 64);
  for (int t = 0; t < Pn; ++t) {
    // gates: dconv([x_t, h]) -> sigmoid
    build_cat0_t<<<egrid0, TB, 0, stream>>>(XDT, XEMBT, HT, t);
    diffuse_t<<<dgrid, 32, 0, stream>>>(Ls, XDT, 0, 1, -1);
    diffuse_t<<<dgrid, 32, 0, stream>>>(Ls, XDT, 1, 2, 0);
    diffuse_t<<<dgrid, 32, 0, stream>>>(Ls, XDT, 2, 3, 1);
    gemm_t<<<dim3(MT / 16, 4), 32, 0, stream>>>(XDT, MT, W_g, 128, b_g, RUT, MT, 512, 2);

    // candidate: dconv([x_t, r*h]) -> tanh
    build_cat1_t<<<egrid1, TB, 0, stream>>>(XDT, RUT, HT);
    diffuse_t<<<dgrid, 32, 0, stream>>>(Ls, XDT, 0, 1, -1);
    diffuse_t<<<dgrid, 32, 0, stream>>>(Ls, XDT, 1, 2, 0);
    diffuse_t<<<dgrid, 32, 0, stream>>>(Ls, XDT, 2, 3, 1);
    gemm_t<<<dim3(MT / 16, 2), 32, 0, stream>>>(XDT, MT, W_c, 64, b_c, CCT, MT, 512, 3);

    // h = u*h + (1-u)*c
    update_h_t<<<egrid1, TB, 0, stream>>>(HT, RUT, CCT);
  }

  // --- output head ---
  gemm_t<<<dim3(MT / 16, 2), 32, 0, stream>>>(HT, MT, W_o1, 64, b_o1, Y1T, MT, 64, 1);
  out_proj_t<<<(Qn * MT + TB - 1) / TB, TB, 0, stream>>>(Y1T, W_o2, b_o2, out);
}